// OracleROUND_64269890617474
// MI455X (gfx1250) — compile-verified
//
#include <hip/hip_runtime.h>
#include <math.h>

#define B_   64
#define S_   2048
#define DIN_ 8
#define H_   512
#define WOBBLE_COUPLING (-1.0f)
#define WOBBLE_STEP     0.03125f

typedef __attribute__((ext_vector_type(2))) float v2f;
typedef __attribute__((ext_vector_type(8))) float v8f;

// Wrap pt into (-pi, pi]: equivalent to atan2(sin(pt), cos(pt)).
// 2-term Cody-Waite so the fold is accurate for |pt| up to ~1e4.
__device__ __forceinline__ float wrap_pi(float pt) {
    const float INV_2PI = 0.15915494309189535f;
    const float C1 = 6.28125f;              // 2*pi split hi
    const float C2 = 1.9353071795864769e-3f; // 2*pi split lo
    float k = rintf(pt * INV_2PI);
    float r = fmaf(-k, C1, pt);
    return fmaf(-k, C2, r);
}

// ---------------------------------------------------------------------------
// Kernel 1: rep[s] = all(x[:, s, :] == x[:, s-1, :]) over the whole batch.
// ---------------------------------------------------------------------------
__global__ void __launch_bounds__(512)
rep_kernel(const float* __restrict__ x, int* __restrict__ rep) {
    __shared__ int flag;
    const int s = blockIdx.x;
    const int t = threadIdx.x;
    if (t == 0) flag = 1;
    __syncthreads();
    if (s == 0) {
        if (t == 0) rep[0] = 0;
        return;
    }
    const int b = t >> 3, d = t & 7;
    const size_t i = ((size_t)b * S_ + s) * DIN_ + d;
    if (x[i] != x[i - DIN_]) flag = 0;   // benign race: all writers store 0
    __syncthreads();
    if (t == 0) rep[s] = flag;
}

// ---------------------------------------------------------------------------
// Kernel 2: WMMA-fed sequential phase scan.
// Grid: (H/256, B). Block: 256 threads = 8 waves; each wave owns 32 h-columns.
// Per 16-step chunk: 4x v_wmma_f32_16x16x4_f32 compute two 16x16 pt tiles
// (K=8 split as 4+4, bias folded into C), spilled through a padded LDS region,
// theta + rep gathered off-chain, then 16 serial steps with coalesced stores.
// ---------------------------------------------------------------------------
__global__ void __launch_bounds__(256)
scan_kernel(const float* __restrict__ x, const float* __restrict__ We,
            const float* __restrict__ be, const int* __restrict__ rep,
            float* __restrict__ ph_hist, float* __restrict__ wb_hist) {
    __shared__ float lds[8][16 * 34];   // row stride 34 floats: conflict-free
    const int lane = threadIdx.x & 31;
    const int w    = threadIdx.x >> 5;
    const int b    = blockIdx.y;
    const int hw   = blockIdx.x * 256 + w * 32;  // wave's 32-column base
    const int h    = hw + lane;                  // this lane's scan column

    const int n  = lane & 15;            // N (column) within a 16-wide tile
    const int m  = lane & 15;            // M (row) for the A operand
    const int kb = (lane >> 4) * 2;      // K pair handled by this lane group

    // B operands (We is [8, H] row-major), constant across the whole scan.
    v2f B0lo, B0hi, B1lo, B1hi;
    B0lo[0] = We[(kb + 0) * H_ + hw + n];      B0lo[1] = We[(kb + 1) * H_ + hw + n];
    B0hi[0] = We[(kb + 4) * H_ + hw + n];      B0hi[1] = We[(kb + 5) * H_ + hw + n];
    B1lo[0] = We[(kb + 0) * H_ + hw + 16 + n]; B1lo[1] = We[(kb + 1) * H_ + hw + 16 + n];
    B1hi[0] = We[(kb + 4) * H_ + hw + 16 + n]; B1hi[1] = We[(kb + 5) * H_ + hw + 16 + n];
    const float be0 = be[hw + n];
    const float be1 = be[hw + 16 + n];

    float ph = 0.0f, wb = 0.0f;
    const size_t hist_bh = (size_t)b * H_ + h;

    for (int s0 = 0; s0 < S_; s0 += 16) {
        // A tile: x[b, s0+m, kb..kb+1] (K 0..3) and kb+4..kb+5 (K 4..7)
        const size_t abase = ((size_t)b * S_ + s0 + m) * DIN_ + kb;
        v2f Alo, Ahi;
        Alo[0] = x[abase + 0];  Alo[1] = x[abase + 1];
        Ahi[0] = x[abase + 4];  Ahi[1] = x[abase + 5];
        if (s0 + 16 < S_)
            __builtin_prefetch(&x[abase + 16 * DIN_], 0, 3);  // global_prefetch_b8

        v8f c0, c1;
#pragma unroll
        for (int v = 0; v < 8; ++v) { c0[v] = be0; c1[v] = be1; }
        // pt = A(16x8) * We(8x16) + be, as two K=4 accumulating WMMAs per tile
        c0 = __builtin_amdgcn_wmma_f32_16x16x4_f32(false, Alo, false, B0lo, (short)0, c0, false, false);
        c0 = __builtin_amdgcn_wmma_f32_16x16x4_f32(false, Ahi, false, B0hi, (short)0, c0, false, false);
        c1 = __builtin_amdgcn_wmma_f32_16x16x4_f32(false, Alo, false, B1lo, (short)0, c1, false, false);
        c1 = __builtin_amdgcn_wmma_f32_16x16x4_f32(false, Ahi, false, B1hi, (short)0, c1, false, false);

        // Spill D tiles: lane l holds rows M = v + 8*(l>>4), col N = l&15.
        const int rbase = (lane >> 4) * 8;
#pragma unroll
        for (int v = 0; v < 8; ++v) {
            const int row = rbase + v;
            lds[w][row * 34 + n]      = c0[v];
            lds[w][row * 34 + 16 + n] = c1[v];
        }

        // Off-chain pass: theta (phase wrap) + rep flags for this chunk.
        float th[16];
        int   rp[16];
#pragma unroll
        for (int si = 0; si < 16; ++si) {
            th[si] = wrap_pi(lds[w][si * 34 + lane]);
            rp[si] = rep[s0 + si];
        }

        // Serial recurrence: 2 dependent hardware sines per step.
#pragma unroll
        for (int si = 0; si < 16; ++si) {
            const float wb1 = wb + WOBBLE_STEP;
            ph = ph + th[si] + WOBBLE_COUPLING * __sinf(wb1);
            const float wbc = wb1 + 0.5f * __sinf(ph - wb1);
            wb = rp[si] ? wbc : wb1;
            const size_t off = (size_t)(s0 + si) * (B_ * H_) + hist_bh;
            ph_hist[off] = ph;
            wb_hist[off] = wb;
        }
    }
}

// ---------------------------------------------------------------------------
// Kernel 3: logits = concat(cos ph, sin ph, cos ph/2, sin ph/2, cos wb,
//                           sin wb, ph) @ Wr + br.   One block per batch row.
// ---------------------------------------------------------------------------
__global__ void __launch_bounds__(256)
readout_kernel(const float* __restrict__ ph_last, const float* __restrict__ wb_last,
               const float* __restrict__ Wr, const float* __restrict__ br,
               float* __restrict__ logits) {
    __shared__ float sm[3][256];
    const int b = blockIdx.x, t = threadIdx.x;
    float a0 = 0.0f, a1 = 0.0f, a2 = 0.0f;
    for (int h = t; h < H_; h += 256) {
        const float p  = ph_last[(size_t)b * H_ + h];
        const float wv = wb_last[(size_t)b * H_ + h];
        const float f[7] = { __cosf(p), __sinf(p), __cosf(0.5f * p), __sinf(0.5f * p),
                             __cosf(wv), __sinf(wv), p };
#pragma unroll
        for (int k = 0; k < 7; ++k) {
            const int j = k * H_ + h;
            a0 += f[k] * Wr[j * 3 + 0];
            a1 += f[k] * Wr[j * 3 + 1];
            a2 += f[k] * Wr[j * 3 + 2];
        }
    }
    sm[0][t] = a0; sm[1][t] = a1; sm[2][t] = a2;
    __syncthreads();
    for (int st = 128; st > 0; st >>= 1) {
        if (t < st) {
            sm[0][t] += sm[0][t + st];
            sm[1][t] += sm[1][t + st];
            sm[2][t] += sm[2][t + st];
        }
        __syncthreads();
    }
    if (t == 0) {
        logits[b * 3 + 0] = sm[0][0] + br[0];
        logits[b * 3 + 1] = sm[1][0] + br[1];
        logits[b * 3 + 2] = sm[2][0] + br[2];
    }
}

// ---------------------------------------------------------------------------
extern "C" void kernel_launch(void* const* d_in, const int* in_sizes, int n_in,
                              void* d_out, int out_size, void* d_ws, size_t ws_size,
                              hipStream_t stream) {
    const float* x  = (const float*)d_in[0];
    const float* We = (const float*)d_in[1];
    const float* be = (const float*)d_in[2];
    const float* Wr = (const float*)d_in[3];
    const float* br = (const float*)d_in[4];

    float* out     = (float*)d_out;
    float* logits  = out;                              // [64, 3]
    float* ph_hist = out + (size_t)B_ * 3;             // [S, B, H]
    float* wb_hist = ph_hist + (size_t)S_ * B_ * H_;   // [S, B, H]
    int*   rep     = (int*)d_ws;                       // [S]

    rep_kernel<<<S_, 512, 0, stream>>>(x, rep);

    dim3 grid(H_ / 256, B_);
    scan_kernel<<<grid, 256, 0, stream>>>(x, We, be, rep, ph_hist, wb_hist);

    const float* ph_last = ph_hist + (size_t)(S_ - 1) * B_ * H_;
    const float* wb_last = wb_hist + (size_t)(S_ - 1) * B_ * H_;
    readout_kernel<<<B_, 256, 0, stream>>>(ph_last, wb_last, Wr, br, logits);
}